// CombinedLoss_dynamic_58085137711777
// MI455X (gfx1250) — compile-verified
//
#include <hip/hip_runtime.h>

// ---------------------------------------------------------------------------
// CombinedLoss_dynamic for MI455X (gfx1250, wave32).
// Single fused pass (HBM-bound, ~268 MB traffic -> ~11.5 us roofline at
// 23.3 TB/s), deterministic two-stage reduction. Final 4096-partial reduction
// uses V_WMMA_F32_16X16X4_F32 (ones B-matrix row-sum trick) so the scalar
// reduce runs on the CDNA5 matrix pipe in full f32.
// ---------------------------------------------------------------------------

typedef __attribute__((ext_vector_type(2))) float v2f;
typedef __attribute__((ext_vector_type(4))) float v4f;
typedef __attribute__((ext_vector_type(8))) float v8f;

#define NBLK 4096
#define NTHR 256

__global__ __launch_bounds__(NTHR) void fused_pde_loss(
    const float* __restrict__ inp,   const float* __restrict__ outc,
    const float* __restrict__ outp,  const float* __restrict__ tcur,
    const float* __restrict__ tpast, const float* __restrict__ tgt,
    float* __restrict__ partials)
{
    constexpr int W = 128, H = 128, Dz = 64;
    constexpr int HW = H * W;                      // 16384
    constexpr int GROUPS = 16 * Dz * H * (W / 4);  // 4,194,304
    const float alpha = 0.0257f;
    const float srcI  = 100000.0f / 27353.34765625f;
    const float thr   = (1000.0f - 20.0f) / 27353.34765625f;

    const int tid = blockIdx.x * NTHR + threadIdx.x;
    const int nth = NBLK * NTHR;
    const v4f zero4 = {};
    float acc = 0.0f;

    for (int g = tid; g < GROUPS; g += nth) {
        const int x = (g & 31) << 2;      // 0..124, step 4
        const int y = (g >> 5) & 127;
        const int z = (g >> 12) & 63;
        const int b = g >> 18;
        const int base = (b << 20) | (z << 14) | (y << 7) | x;

        // Stencil center + neighbors. Input is reused 7x -> keep it resident
        // in L2 (67 MB tensor vs 192 MB L2); neighbor rows are L2 hits.
        const v4f c = *(const v4f*)(inp + base);
        const float xl = (x > 0)     ? inp[base - 1] : 0.0f;
        const float xr = (x < W - 4) ? inp[base + 4] : 0.0f;
        v4f ym = zero4, yp = zero4, zm = zero4, zp = zero4;
        if (y > 0)      ym = *(const v4f*)(inp + base - W);
        if (y < H - 1)  yp = *(const v4f*)(inp + base + W);
        if (z > 0)      zm = *(const v4f*)(inp + base - HW);
        if (z < Dz - 1) zp = *(const v4f*)(inp + base + HW);

        // Single-use streams: non-temporal so they don't evict `inp` from L2.
        const v4f o = __builtin_nontemporal_load((const v4f*)(outc + base));
        const v4f p = __builtin_nontemporal_load((const v4f*)(outp + base));
        const v4f q = __builtin_nontemporal_load((const v4f*)(tgt  + base));

        const float idt = 1.0f / (tcur[b] - tpast[b]);

        const float lap0 = xl  + c.y + ym.x + yp.x + zm.x + zp.x - 6.0f * c.x;
        const float lap1 = c.x + c.z + ym.y + yp.y + zm.y + zp.y - 6.0f * c.y;
        const float lap2 = c.y + c.w + ym.z + yp.z + zm.z + zp.z - 6.0f * c.z;
        const float lap3 = c.z + xr  + ym.w + yp.w + zm.w + zp.w - 6.0f * c.w;

        const float r0 = (o.x - p.x) * idt - alpha * lap0 - ((c.x > thr) ? srcI : 0.0f);
        const float r1 = (o.y - p.y) * idt - alpha * lap1 - ((c.y > thr) ? srcI : 0.0f);
        const float r2 = (o.z - p.z) * idt - alpha * lap2 - ((c.z > thr) ? srcI : 0.0f);
        const float r3 = (o.w - p.w) * idt - alpha * lap3 - ((c.w > thr) ? srcI : 0.0f);

        const float e0 = o.x - q.x, e1 = o.y - q.y, e2 = o.z - q.z, e3 = o.w - q.w;

        acc = fmaf(r0, r0, acc); acc = fmaf(e0, e0, acc);
        acc = fmaf(r1, r1, acc); acc = fmaf(e1, e1, acc);
        acc = fmaf(r2, r2, acc); acc = fmaf(e2, e2, acc);
        acc = fmaf(r3, r3, acc); acc = fmaf(e3, e3, acc);
    }

    // wave32 reduction
    for (int off = 16; off > 0; off >>= 1)
        acc += __shfl_down(acc, off, 32);

    __shared__ float smem[NTHR / 32];
    const int wid = threadIdx.x >> 5;
    if ((threadIdx.x & 31) == 0) smem[wid] = acc;
    __syncthreads();
    if (threadIdx.x == 0) {
        float s = 0.0f;
        #pragma unroll
        for (int i = 0; i < NTHR / 32; ++i) s += smem[i];
        partials[blockIdx.x] = s;  // deterministic: fixed tree, no atomics
    }
}

// ---------------------------------------------------------------------------
// Final reduction of NBLK (=4096) partials with V_WMMA_F32_16X16X4_F32.
// B = ones(4x16)  =>  D[m][n] = sum_k A[m][k] + C[m][n]; any bijection of 64
// loaded values onto A's slots preserves the total. After 64 chained WMMAs,
// column n=0 of C holds the 16 row totals: lane 0 carries M=0..7 in VGPRs
// 0..7, lane 16 carries M=8..15 (ISA 16x16 f32 C/D layout).
// One wave => EXEC all ones, as WMMA requires.
// ---------------------------------------------------------------------------
__global__ __launch_bounds__(32) void reduce_partials_wmma(
    const float* __restrict__ partials, float* __restrict__ out, float inv_n)
{
    const int lane = threadIdx.x;  // 0..31
    v8f acc = {};
    v2f ones; ones.x = 1.0f; ones.y = 1.0f;

    #pragma unroll 4
    for (int i = 0; i < NBLK / 64; ++i) {
        v2f a;
        a.x = partials[i * 64 + lane];
        a.y = partials[i * 64 + 32 + lane];
        // 8 args: (neg_a, A, neg_b, B, c_mod, C, reuse_a, reuse_b)
        acc = __builtin_amdgcn_wmma_f32_16x16x4_f32(
            false, a, false, ones, (short)0, acc, false, false);
    }

    float s = acc[0] + acc[1] + acc[2] + acc[3] +
              acc[4] + acc[5] + acc[6] + acc[7];       // sum of my column
    const float total = __shfl(s, 0, 32) + __shfl(s, 16, 32);  // column n=0
    if (lane == 0) out[0] = total * inv_n;
}

extern "C" void kernel_launch(void* const* d_in, const int* in_sizes, int n_in,
                              void* d_out, int out_size, void* d_ws, size_t ws_size,
                              hipStream_t stream)
{
    const float* inp   = (const float*)d_in[0];
    const float* outc  = (const float*)d_in[1];
    const float* outp  = (const float*)d_in[2];
    const float* tcur  = (const float*)d_in[3];
    const float* tpast = (const float*)d_in[4];
    const float* tgt   = (const float*)d_in[5];
    float* partials = (float*)d_ws;       // NBLK floats = 16 KB scratch
    float* out = (float*)d_out;

    fused_pde_loss<<<NBLK, NTHR, 0, stream>>>(inp, outc, outp, tcur, tpast, tgt,
                                              partials);

    const float inv_n = 1.0f / (16.0f * 64.0f * 128.0f * 128.0f);
    reduce_partials_wmma<<<1, 32, 0, stream>>>(partials, out, inv_n);
}